// ChannelCollator_38500086841890
// MI455X (gfx1250) — compile-verified
//
#include <hip/hip_runtime.h>
#include <math.h>

// ---------------- problem geometry ----------------
#define B_DIM 128
#define T_DIM 16384
#define P_DIM 20
#define C_DIM 18
#define BC    (B_DIM * C_DIM)      // 2304 channels
#define L_BLK 512                  // samples per scan block
#define NB    (T_DIM / L_BLK)      // 32 blocks per channel
#define NINST (BC * NB)            // 73728 block instances

typedef float v2f __attribute__((ext_vector_type(2)));
typedef float v8f __attribute__((ext_vector_type(8)));

struct Coeffs {
  float hb0, hb1, hb2, ha1, ha2;   // highpass biquad (normalized)
  float lb0, lb1, lb2, la1, la2;   // lowpass  biquad (normalized)
};

// bipolar montage pair indices (dict order LL, LP, RP, RL, Z)
__constant__ int c_P1[C_DIM] = {0,4,5,6, 0,1,2,3, 11,15,16,17, 11,12,13,14, 8,9};
__constant__ int c_P2[C_DIM] = {4,5,6,7, 1,2,3,7, 15,16,17,18, 12,13,14,18, 9,10};

// ---------------------------------------------------------------------------
// Stage A: homogeneous response H (L x 4) and 4x4 block transition M.
// Column k = response to unit initial state e_k, k in {v1,v2,y1,y2}, x == 0.
// ---------------------------------------------------------------------------
__global__ void hm_kernel(float* __restrict__ H, float* __restrict__ M, Coeffs cf) {
  int k = threadIdx.x;
  if (k >= 4) return;
  float v1 = (k == 0) ? 1.f : 0.f;
  float v2 = (k == 1) ? 1.f : 0.f;
  float y1 = (k == 2) ? 1.f : 0.f;
  float y2 = (k == 3) ? 1.f : 0.f;
  for (int n = 0; n < L_BLK; ++n) {
    float v = -cf.ha1 * v1 - cf.ha2 * v2;
    float y = cf.lb0 * v + cf.lb1 * v1 + cf.lb2 * v2 - cf.la1 * y1 - cf.la2 * y2;
    H[n * 4 + k] = y;
    v2 = v1; v1 = v; y2 = y1; y1 = y;
  }
  M[0 * 4 + k] = v1;  // state row i responds to incoming state k
  M[1 * 4 + k] = v2;
  M[2 * 4 + k] = y1;
  M[3 * 4 + k] = y2;
}

// ---------------------------------------------------------------------------
// Stage B (fused): per-(channel, block) zero-state cascade + pair-mask output.
// Thread map: c fastest -> consecutive runs of 18 lanes share (b, blk) and read the same
// 80-byte probe row each step (coalesced); per-lane output streams are
// contiguous in n, written as float4 (B128) every 4 samples.
// ---------------------------------------------------------------------------
__device__ __forceinline__ void masked_diff(const float* __restrict__ xb,
                                            const float* __restrict__ mb,
                                            int t, int p1, int p2,
                                            float& xd, float& m) {
  int o = t * P_DIM;
  m  = mb[o + p1] * mb[o + p2];
  xd = (xb[o + p1] - xb[o + p2]) * m;
}

__global__ void __launch_bounds__(256)
block_filter_kernel(const float* __restrict__ x,
                    const float* __restrict__ mask,
                    float* __restrict__ out_eegs,
                    float* __restrict__ out_masks,
                    float* __restrict__ pbuf, Coeffs cf) {
  int tid = blockIdx.x * blockDim.x + threadIdx.x;  // [0, NINST)
  int c   = tid % C_DIM;                            // fastest: coalesced lanes
  int rem = tid / C_DIM;
  int blk = rem % NB;
  int b   = rem / NB;
  int ch  = b * C_DIM + c;
  int p1 = c_P1[c], p2 = c_P2[c];
  const float* xb = x    + (long long)b * T_DIM * P_DIM;
  const float* mb = mask + (long long)b * T_DIM * P_DIM;
  int t0 = blk * L_BLK;

  float x1 = 0.f, x2 = 0.f, dm;
  if (blk > 0) {
    masked_diff(xb, mb, t0 - 1, p1, p2, x1, dm);
    masked_diff(xb, mb, t0 - 2, p1, p2, x2, dm);
  }
  float v1 = 0.f, v2 = 0.f, y1 = 0.f, y2 = 0.f;

  float* orow = out_eegs  + (long long)ch * T_DIM + t0;
  float* mrow = out_masks + (long long)ch * T_DIM + t0;
  for (int n = 0; n < L_BLK; n += 4) {
    // stream ahead (lanes share the line; cache dedups)
    __builtin_prefetch(xb + (t0 + n + 32) * P_DIM, 0, 0);
    __builtin_prefetch(mb + (t0 + n + 32) * P_DIM, 0, 0);
    float4 yq, mq;
#pragma unroll
    for (int u = 0; u < 4; ++u) {
      float xd, m;
      masked_diff(xb, mb, t0 + n + u, p1, p2, xd, m);
      float v = cf.hb0 * xd + cf.hb1 * x1 + cf.hb2 * x2 - cf.ha1 * v1 - cf.ha2 * v2;
      float y = cf.lb0 * v  + cf.lb1 * v1 + cf.lb2 * v2 - cf.la1 * y1 - cf.la2 * y2;
      (&yq.x)[u] = y;
      (&mq.x)[u] = m;
      x2 = x1; x1 = xd; v2 = v1; v1 = v; y2 = y1; y1 = y;
    }
    *(float4*)(orow + n) = yq;   // 16B-aligned: t0, n multiples of 4
    *(float4*)(mrow + n) = mq;
  }
  float* p = pbuf + (long long)(ch * NB + blk) * 4;
  p[0] = v1; p[1] = v2; p[2] = y1; p[3] = y2;
}

// ---------------------------------------------------------------------------
// Stage C: sequential 32-step scan per channel: s_{j+1} = M s_j + p_j.
// Stores the *incoming* state of every block into sbuf.
// ---------------------------------------------------------------------------
__global__ void scan_kernel(const float* __restrict__ Mm,
                            const float* __restrict__ pbuf,
                            float* __restrict__ sbuf) {
  int ch = blockIdx.x * blockDim.x + threadIdx.x;
  if (ch >= BC) return;
  float m[16];
#pragma unroll
  for (int i = 0; i < 16; ++i) m[i] = Mm[i];
  float s0 = 0.f, s1 = 0.f, s2 = 0.f, s3 = 0.f;
  for (int j = 0; j < NB; ++j) {
    long long base = ((long long)ch * NB + j) * 4;
    sbuf[base + 0] = s0; sbuf[base + 1] = s1;
    sbuf[base + 2] = s2; sbuf[base + 3] = s3;
    float t0 = m[0]  * s0 + m[1]  * s1 + m[2]  * s2 + m[3]  * s3 + pbuf[base + 0];
    float t1 = m[4]  * s0 + m[5]  * s1 + m[6]  * s2 + m[7]  * s3 + pbuf[base + 1];
    float t2 = m[8]  * s0 + m[9]  * s1 + m[10] * s2 + m[11] * s3 + pbuf[base + 2];
    float t3 = m[12] * s0 + m[13] * s1 + m[14] * s2 + m[15] * s3 + pbuf[base + 3];
    s0 = t0; s1 = t1; s2 = t2; s3 = t3;
  }
}

// ---------------------------------------------------------------------------
// Stage D: WMMA correction.  D = H_tile(16x4) x S(4x16) + C(current output).
// One wave per 16-sample x 16-instance tile; full waves, EXEC all ones.
//   A (16x4 f32): lane M = lane&15; VGPRs hold K={0,1} (lanes 0-15) /
//     K={2,3} (lanes 16-31).
//   B (4x16 f32): lane N = lane&15; same K split across lane halves.
//   C/D (16x16 f32): VGPR r holds row M = r + 8*(lane>=16), col N = lane&15.
// ---------------------------------------------------------------------------
__global__ void __launch_bounds__(256)
wmma_correct_kernel(const float* __restrict__ H,
                    const float* __restrict__ sbuf,
                    float* __restrict__ out_eegs) {
  const int lane = threadIdx.x & 31;
  const int wave = threadIdx.x >> 5;
  const int tile = blockIdx.x * 8 + wave;          // [0, 147456)
  const int mt      = tile & (L_BLK / 16 - 1);     // sample-tile within block
  const int ci_base = (tile >> 5) * 16;            // first column instance

  const int nsub = lane & 15;
  const int half = lane >> 4;

  v2f a;   // A: rows of H for samples mt*16 .. mt*16+15
  {
    const float* hrow = H + (mt * 16 + nsub) * 4 + half * 2;
    a.x = hrow[0];
    a.y = hrow[1];
  }
  v2f bm;  // B: incoming states for the 16 instances of this column tile
  {
    const float* srow = sbuf + (long long)(ci_base + nsub) * 4 + half * 2;
    bm.x = srow[0];
    bm.y = srow[1];
  }
  // C: current zero-state outputs (accumulate correction onto them)
  const int ci  = ci_base + nsub;                  // NB=32 -> never crosses a channel
  const int ch  = ci / NB;
  const int blk = ci % NB;
  float* col = out_eegs + (long long)ch * T_DIM + blk * L_BLK + mt * 16 + half * 8;
  v8f cm;
#pragma unroll
  for (int r = 0; r < 8; ++r) cm[r] = col[r];

  v8f d = __builtin_amdgcn_wmma_f32_16x16x4_f32(
      /*neg_a=*/false, a, /*neg_b=*/false, bm,
      /*c_mod=*/(short)0, cm, /*reuse_a=*/false, /*reuse_b=*/false);

#pragma unroll
  for (int r = 0; r < 8; ++r) col[r] = d[r];
}

// ---------------------------------------------------------------------------
extern "C" void kernel_launch(void* const* d_in, const int* in_sizes, int n_in,
                              void* d_out, int out_size, void* d_ws, size_t ws_size,
                              hipStream_t stream) {
  (void)in_sizes; (void)n_in; (void)out_size; (void)ws_size;
  const float* x    = (const float*)d_in[0];
  const float* mask = (const float*)d_in[1];
  float* out_eegs  = (float*)d_out;
  float* out_masks = out_eegs + (long long)BC * T_DIM;

  float* wsf  = (float*)d_ws;
  float* H    = wsf;                        // 2048 floats
  float* Mm   = wsf + 2048;                 // 16 floats
  float* pbuf = wsf + 2176;                 // NINST*4 floats
  float* sbuf = pbuf + (long long)NINST * 4;// NINST*4 floats  (~2.4 MB total)

  // Host-side coefficient math (pure CPU, graph-capture safe).
  const double q = 0.7071067811865476, sr = 40.0, PI = 3.14159265358979323846;
  Coeffs cf;
  {
    double w0 = 2.0 * PI * 0.5 / sr;
    double al = sin(w0) / (2.0 * q), co = cos(w0), a0 = 1.0 + al;
    cf.hb0 = (float)(((1.0 + co) * 0.5) / a0);
    cf.hb1 = (float)((-(1.0 + co)) / a0);
    cf.hb2 = (float)(((1.0 + co) * 0.5) / a0);
    cf.ha1 = (float)((-2.0 * co) / a0);
    cf.ha2 = (float)((1.0 - al) / a0);
  }
  {
    double w0 = 2.0 * PI * 50.0 / sr;
    double al = sin(w0) / (2.0 * q), co = cos(w0), a0 = 1.0 + al;
    cf.lb0 = (float)(((1.0 - co) * 0.5) / a0);
    cf.lb1 = (float)((1.0 - co) / a0);
    cf.lb2 = (float)(((1.0 - co) * 0.5) / a0);
    cf.la1 = (float)((-2.0 * co) / a0);
    cf.la2 = (float)((1.0 - al) / a0);
  }

  hm_kernel<<<1, 32, 0, stream>>>(H, Mm, cf);
  block_filter_kernel<<<NINST / 256, 256, 0, stream>>>(x, mask, out_eegs,
                                                       out_masks, pbuf, cf);
  scan_kernel<<<(BC + 255) / 256, 256, 0, stream>>>(Mm, pbuf, sbuf);
  const int total_tiles = (NINST / 16) * (L_BLK / 16);   // 147456 wave-tiles
  wmma_correct_kernel<<<total_tiles / 8, 256, 0, stream>>>(H, sbuf, out_eegs);
}